// CrossLayerSparseMoE_64141041598886
// MI455X (gfx1250) — compile-verified
//
#include <hip/hip_runtime.h>
#include <hip/hip_bf16.h>
#include <math.h>

// ---------------------------------------------------------------------------
// CrossLayer Sparse MoE for MI455X (gfx1250, wave32).
// f32 reference semantics preserved via V_WMMA_F32_16X16X4_F32.
// Pipeline: zero_counter -> router -> capacity scan -> init_out -> expert FFN.
// ---------------------------------------------------------------------------

#define D_DIM    512
#define E_DIM    8
#define H_DIM    2048
#define TOPK     2
#define CAPMAX   8192          // max capacity = floor(N*TOPK/E) = 8192
#define MT       32            // tokens per expert block
#define HC       64            // H chunk per fused-FFN iteration
#define XS_STRIDE 516          // 516 % 64 == 4 -> conflict-free A-frag b64 reads
#define HS_STRIDE 68           // 68  % 64 == 4 -> same trick for the h tile

typedef float v2f __attribute__((ext_vector_type(2)));
typedef float v8f __attribute__((ext_vector_type(8)));

// ---------------------------------------------------------------------------
__global__ void zero_counter_kernel(unsigned* counter) {
    if (threadIdx.x == 0) *counter = 0u;
}

// ---------------------------------------------------------------------------
// Router: one wave per token. Coalesced x reads; Wr/Wn/Ws are tiny (L1/L2
// resident). Lane-parallel partial dots, shfl tree reduction, lane 0 does the
// scalar top-2 / softmax / skip logic.
// ---------------------------------------------------------------------------
__global__ void __launch_bounds__(256)
router_kernel(const float* __restrict__ x, const float* __restrict__ noise,
              const float* __restrict__ Wr, const float* __restrict__ br,
              const float* __restrict__ Wn, const float* __restrict__ bn,
              const float* __restrict__ Ws, const float* __restrict__ bs,
              int* __restrict__ mask, float* __restrict__ gates,
              unsigned* __restrict__ counter, int N) {
    const int wave = (int)((blockIdx.x * blockDim.x + threadIdx.x) >> 5);
    const int lane = threadIdx.x & 31;
    if (wave >= N) return;

    const float* xr = x + (size_t)wave * D_DIM;
    float lr[E_DIM], ln[E_DIM];
#pragma unroll
    for (int e = 0; e < E_DIM; ++e) { lr[e] = 0.f; ln[e] = 0.f; }
    float ls = 0.f;

#pragma unroll
    for (int j = 0; j < D_DIM / 32; ++j) {
        const int d = j * 32 + lane;
        const float xv = xr[d];
        const float* wr = Wr + d * E_DIM;
        const float* wn = Wn + d * E_DIM;
#pragma unroll
        for (int e = 0; e < E_DIM; ++e) {
            lr[e] = fmaf(xv, wr[e], lr[e]);
            ln[e] = fmaf(xv, wn[e], ln[e]);
        }
        ls = fmaf(xv, Ws[d], ls);
    }
    // wave32 tree reduction
#pragma unroll
    for (int off = 16; off > 0; off >>= 1) {
#pragma unroll
        for (int e = 0; e < E_DIM; ++e) {
            lr[e] += __shfl_down(lr[e], off, 32);
            ln[e] += __shfl_down(ln[e], off, 32);
        }
        ls += __shfl_down(ls, off, 32);
    }

    if (lane == 0) {
        float noisy[E_DIM];
#pragma unroll
        for (int e = 0; e < E_DIM; ++e) {
            const float lg = lr[e] + br[e];
            const float nl = ln[e] + bn[e];
            float sp;                           // softplus, stable
            if (nl > 20.f)       sp = nl;
            else if (nl < -20.f) sp = expf(nl);
            else                 sp = log1pf(expf(nl));
            noisy[e] = fmaf(noise[(size_t)wave * E_DIM + e], sp, lg);
        }
        // top-2, first index wins ties (matches jax.lax.top_k)
        int i1 = 0;
#pragma unroll
        for (int e = 1; e < E_DIM; ++e) if (noisy[e] > noisy[i1]) i1 = e;
        int i2 = (i1 == 0) ? 1 : 0;
#pragma unroll
        for (int e = 0; e < E_DIM; ++e)
            if (e != i1 && noisy[e] > noisy[i2]) i2 = e;

        const bool nonskip = !((ls + bs[0]) > 0.0f);
        const float m  = fmaxf(noisy[i1], noisy[i2]);
        const float e1 = expf(noisy[i1] - m);
        const float e2 = expf(noisy[i2] - m);
        const float inv = 1.0f / (e1 + e2);

        float g[E_DIM];
#pragma unroll
        for (int e = 0; e < E_DIM; ++e) g[e] = 0.f;
        g[i1] = e1 * inv;
        g[i2] = e2 * inv;
#pragma unroll
        for (int e = 0; e < E_DIM; ++e) gates[(size_t)wave * E_DIM + e] = g[e];

        mask[wave] = nonskip ? ((1 << i1) | (1 << i2) | 0x100) : 0;
        if (nonskip) atomicAdd(counter, 1u);
    }
}

// ---------------------------------------------------------------------------
// Capacity scan: single 1024-thread workgroup, sequential tiles to preserve
// the reference's flat-order per-expert prefix ranks. Emits compacted
// per-expert (token, gate) lists of length min(total, capacity).
// ---------------------------------------------------------------------------
__global__ void __launch_bounds__(1024)
scan_kernel(const int* __restrict__ mask, const float* __restrict__ gates,
            const unsigned* __restrict__ counter,
            int* __restrict__ expIdx, float* __restrict__ expGate,
            int* __restrict__ counts, int N) {
    __shared__ int waveTot[32][E_DIM];
    __shared__ int waveBase[32][E_DIM];
    __shared__ int blockBase[E_DIM];
    __shared__ int tileTot[E_DIM];

    const int t = threadIdx.x;
    const int lane = t & 31;
    const int wv = t >> 5;
    const int capacity = (int)((*counter) * (unsigned)TOPK / (unsigned)E_DIM);

    if (t < E_DIM) blockBase[t] = 0;
    __syncthreads();

    const unsigned long long lt = ((1ull << lane) - 1ull);
    for (int tile = 0; tile < N; tile += 1024) {
        const int n = tile + t;
        const int m = (n < N) ? mask[n] : 0;
        int pre[E_DIM];
#pragma unroll
        for (int e = 0; e < E_DIM; ++e) {
            const int flag = (m >> e) & 1;
            const unsigned long long b = __ballot(flag);
            pre[e] = __popcll(b & lt);
            if (lane == 0) waveTot[wv][e] = __popcll(b);
        }
        __syncthreads();
        if (t < E_DIM) {                 // serial scan of 32 wave totals
            int run = 0;
            for (int w = 0; w < 32; ++w) { waveBase[w][t] = run; run += waveTot[w][t]; }
            tileTot[t] = run;
        }
        __syncthreads();
#pragma unroll
        for (int e = 0; e < E_DIM; ++e) {
            if ((m >> e) & 1) {
                const int r = blockBase[e] + waveBase[wv][e] + pre[e];
                if (r < capacity) {
                    expIdx[e * CAPMAX + r]  = n;
                    expGate[e * CAPMAX + r] = gates[(size_t)n * E_DIM + e];
                }
            }
        }
        __syncthreads();
        if (t < E_DIM) blockBase[t] += tileTot[t];
        __syncthreads();
    }
    if (t < E_DIM) counts[t] = min(blockBase[t], capacity);
}

// ---------------------------------------------------------------------------
// Output base: skipped tokens pass x through; non-skipped start at 0 and
// receive gate-weighted expert contributions (capacity-dropped tokens stay 0,
// matching the reference).
// ---------------------------------------------------------------------------
__global__ void __launch_bounds__(256)
init_out_kernel(const float* __restrict__ x, const int* __restrict__ mask,
                float* __restrict__ out, int N) {
    const long long tid = (long long)blockIdx.x * blockDim.x + threadIdx.x;
    const long long total = (long long)N * (D_DIM / 4);
    if (tid >= total) return;
    const int n = (int)(tid >> 7);           // D/4 = 128 float4 per row
    const int c = (int)(tid & 127);
    const bool nonskip = (mask[n] & 0x100) != 0;
    float4 v;
    if (nonskip) {
        v = make_float4(0.f, 0.f, 0.f, 0.f);
    } else {
        v = ((const float4*)(x + (size_t)n * D_DIM))[c];
    }
    ((float4*)(out + (size_t)n * D_DIM))[c] = v;
}

// ---------------------------------------------------------------------------
// Fused expert FFN: block = (expert, 32-token tile), 8 waves in a 2x4
// (row-half x D-quarter) grid. h = relu(X@W1+b1) staged through LDS in
// 64-wide H chunks; out = h@W2 accumulated in 8 v8f WMMA accumulators per
// wave; epilogue scatters gate*(out+b2) with f32 atomics.
// ---------------------------------------------------------------------------
__global__ void __launch_bounds__(256)
expert_kernel(const float* __restrict__ x,
              const float* __restrict__ W1, const float* __restrict__ b1,
              const float* __restrict__ W2, const float* __restrict__ b2,
              const int* __restrict__ expIdx, const float* __restrict__ expGate,
              const int* __restrict__ counts, float* __restrict__ out) {
    __shared__ float Xs[MT * XS_STRIDE];
    __shared__ float Hs[MT * HS_STRIDE];
    __shared__ int   sIdx[MT];
    __shared__ float sGate[MT];

    const int e = blockIdx.y;
    const int count = counts[e];
    const int m0 = blockIdx.x * MT;
    if (m0 >= count) return;

    const int tid  = threadIdx.x;
    const int lane = tid & 31;
    const int wv   = tid >> 5;
    const int mh   = wv >> 2;              // row half (0..1)
    const int dq   = wv & 3;               // output-D quarter (0..3)
    const int lcol = lane & 15;            // N index within 16x16 tile
    const int rsel = (lane >> 4) << 1;     // K sub-select for A/B fragments
    const int radd = (lane >> 4) << 3;     // +8 row offset for upper lane half

    const float* W1e = W1 + (size_t)e * D_DIM * H_DIM;
    const float* W2e = W2 + (size_t)e * H_DIM * D_DIM;
    const float* b1e = b1 + (size_t)e * H_DIM;
    const float* b2e = b2 + (size_t)e * D_DIM;

    if (tid < MT) {
        const int g = m0 + tid;
        const bool ok = g < count;
        sIdx[tid]  = ok ? expIdx[e * CAPMAX + g] : 0;
        sGate[tid] = ok ? expGate[e * CAPMAX + g] : 0.0f;
    }
    __syncthreads();

    // Gather X tile [MT x 512] -> LDS (float4, 16 per thread)
#pragma unroll
    for (int i = 0; i < (MT * D_DIM / 4) / 256; ++i) {
        const int f   = tid + i * 256;
        const int row = f >> 7;            // 128 float4 per row
        const int c4  = f & 127;
        const float4 v = ((const float4*)(x + (size_t)sIdx[row] * D_DIM))[c4];
        *(float4*)(&Xs[row * XS_STRIDE + c4 * 4]) = v;
    }
    __syncthreads();

    v8f acc[8];
#pragma unroll
    for (int nt = 0; nt < 8; ++nt) acc[nt] = (v8f){0.f,0.f,0.f,0.f,0.f,0.f,0.f,0.f};

    const float* Arow = &Xs[(mh * 16 + lcol) * XS_STRIDE + rsel];
    const float* Hrow = &Hs[(mh * 16 + lcol) * HS_STRIDE + rsel];

#pragma unroll 1
    for (int hc = 0; hc < H_DIM / HC; ++hc) {
        // ---- GEMM1: 16x16 h tile per wave, K = 512 in steps of 4 ----
        v8f hacc = (v8f){0.f,0.f,0.f,0.f,0.f,0.f,0.f,0.f};
        const int hcol = hc * HC + dq * 16 + lcol;     // column in W1/H
#pragma unroll 8
        for (int k0 = 0; k0 < D_DIM; k0 += 4) {
            const v2f a = *(const v2f*)(Arow + k0);
            v2f b;
            b.x = W1e[(size_t)(k0 + rsel)     * H_DIM + hcol];
            b.y = W1e[(size_t)(k0 + rsel + 1) * H_DIM + hcol];
            hacc = __builtin_amdgcn_wmma_f32_16x16x4_f32(
                       false, a, false, b, (short)0, hacc, false, false);
        }
        // bias + relu, stage into LDS h tile
        {
            const float bias = b1e[hcol];
            const int hcl = dq * 16 + lcol;
#pragma unroll
            for (int r = 0; r < 8; ++r) {
                const int row = mh * 16 + r + radd;
                Hs[row * HS_STRIDE + hcl] = fmaxf(hacc[r] + bias, 0.0f);
            }
        }
        __syncthreads();

        // ---- GEMM2: acc[16 x 128] += h[16 x 64] @ W2[64 x 128] ----
#pragma unroll 2
        for (int k0 = 0; k0 < HC; k0 += 4) {
            const v2f a = *(const v2f*)(Hrow + k0);
            const size_t kg = (size_t)(hc * HC + k0 + rsel) * D_DIM;
#pragma unroll
            for (int nt = 0; nt < 8; ++nt) {
                const int col = dq * 128 + nt * 16 + lcol;
                v2f b;
                b.x = W2e[kg + col];
                b.y = W2e[kg + D_DIM + col];
                acc[nt] = __builtin_amdgcn_wmma_f32_16x16x4_f32(
                              false, a, false, b, (short)0, acc[nt], false, false);
            }
        }
        __syncthreads();
    }

    // ---- epilogue: gate * (out + b2) scattered with f32 atomics ----
#pragma unroll
    for (int nt = 0; nt < 8; ++nt) {
        const int col = dq * 128 + nt * 16 + lcol;
        const float bias = b2e[col];
#pragma unroll
        for (int r = 0; r < 8; ++r) {
            const int row = mh * 16 + r + radd;
            const float v = sGate[row] * (acc[nt][r] + bias);
            atomicAdd(&out[(size_t)sIdx[row] * D_DIM + col], v);
        }
    }
}

// ---------------------------------------------------------------------------
extern "C" void kernel_launch(void* const* d_in, const int* in_sizes, int n_in,
                              void* d_out, int out_size, void* d_ws, size_t ws_size,
                              hipStream_t stream) {
    const float* x     = (const float*)d_in[0];
    const float* noise = (const float*)d_in[1];
    const float* Wr    = (const float*)d_in[2];
    const float* br    = (const float*)d_in[3];
    const float* Wn    = (const float*)d_in[4];
    const float* bn    = (const float*)d_in[5];
    const float* Ws    = (const float*)d_in[6];
    const float* bs    = (const float*)d_in[7];
    const float* W1    = (const float*)d_in[8];
    const float* b1    = (const float*)d_in[9];
    const float* W2    = (const float*)d_in[10];
    const float* b2    = (const float*)d_in[11];
    float* out = (float*)d_out;

    const int N = in_sizes[0] / D_DIM;     // B*S tokens

    // workspace layout
    char* ws = (char*)d_ws;
    unsigned* counter = (unsigned*)ws;                                   // 4 B
    int*      counts  = (int*)(ws + 64);                                 // 32 B
    int*      mask    = (int*)(ws + 128);                                // 4N
    float*    gates   = (float*)(ws + 128 + (size_t)N * 4);              // 32N
    int*      expIdx  = (int*)(ws + 128 + (size_t)N * 36);               // 256 KB
    float*    expGate = (float*)(ws + 128 + (size_t)N * 36 +
                                 (size_t)E_DIM * CAPMAX * 4);            // 256 KB

    zero_counter_kernel<<<1, 32, 0, stream>>>(counter);

    {   // one wave per token
        const int waves_per_block = 256 / 32;
        const int blocks = (N + waves_per_block - 1) / waves_per_block;
        router_kernel<<<blocks, 256, 0, stream>>>(x, noise, Wr, br, Wn, bn,
                                                  Ws, bs, mask, gates, counter, N);
    }

    scan_kernel<<<1, 1024, 0, stream>>>(mask, gates, counter,
                                        expIdx, expGate, counts, N);

    {   // out = nonskip ? 0 : x
        const long long total = (long long)N * (D_DIM / 4);
        const int blocks = (int)((total + 255) / 256);
        init_out_kernel<<<blocks, 256, 0, stream>>>(x, mask, out, N);
    }

    {   // fused expert FFN: x = token tiles (max capacity), y = experts
        dim3 grid(CAPMAX / MT, E_DIM);
        expert_kernel<<<grid, 256, 0, stream>>>(x, W1, b1, W2, b2,
                                                expIdx, expGate, counts, out);
    }
}